// Pretrain_loss_train_22299470201298
// MI455X (gfx1250) — compile-verified
//
#include <hip/hip_runtime.h>

// ---------------------------------------------------------------------------
// CDNA5 (gfx1250) implementation.
// GEMMs use V_WMMA_F32_16X16X4_F32 (fp32 in/out, wave32, 16 chained WMMAs
// cover K=64). Workload is HBM-bound (~300-400 MB moved, ~4.3 GFLOP), so fp32
// matrix math is free and preserves the sort/compare semantics downstream.
// ---------------------------------------------------------------------------

typedef __attribute__((ext_vector_type(2))) float v2f;
typedef __attribute__((ext_vector_type(8))) float v8f;

__device__ __forceinline__ v8f wmma_f32_16x16x4(v2f a, v2f b, v8f c) {
    // 8-arg form: (neg_a, A, neg_b, B, c_mod, C, reuse_a, reuse_b)
    return __builtin_amdgcn_wmma_f32_16x16x4_f32(false, a, false, b,
                                                 (short)0, c, false, false);
}

// ---------------------------------------------------------------------------
// C[M,N] = A[M,64] * B[N,64]^T   (row-major, fp32)
// One wave per 16x16 tile; A/B fragments per ISA layout:
//   lane 0-15  -> rows, K pair {0,1}; lane 16-31 -> rows, K pair {2,3}
// Edge tiles: clamped loads (EXEC stays all-ones through WMMA), guarded store.
// ---------------------------------------------------------------------------
__global__ __launch_bounds__(256)
void gemm_nt_wmma(const float* __restrict__ A, const float* __restrict__ B,
                  float* __restrict__ C, int M, int N, int ldc) {
    const int wave = threadIdx.x >> 5;
    const int lane = threadIdx.x & 31;
    const int tilesN = (N + 15) >> 4;
    const int tilesM = (M + 15) >> 4;
    int tile = blockIdx.x * 8 + wave;
    if (tile >= tilesM * tilesN) return;           // wave-uniform exit
    int tm = tile / tilesN, tn = tile - tm * tilesN;
    int half = lane >> 4, l15 = lane & 15;
    int arow = tm * 16 + l15; if (arow > M - 1) arow = M - 1;
    int brow = tn * 16 + l15; if (brow > N - 1) brow = N - 1;
    const float* Ap = A + (size_t)arow * 64 + half * 2;
    const float* Bp = B + (size_t)brow * 64 + half * 2;
    v8f acc = {};
#pragma unroll
    for (int k = 0; k < 64; k += 4) {
        v2f a; a.x = Ap[k]; a.y = Ap[k + 1];
        v2f b; b.x = Bp[k]; b.y = Bp[k + 1];
        acc = wmma_f32_16x16x4(a, b, acc);
    }
    int col = tn * 16 + l15;
    if (col < N) {
#pragma unroll
        for (int i = 0; i < 8; ++i) {
            int row = tm * 16 + half * 8 + i;      // C layout: VGPR i -> M=i (+8 for hi half)
            if (row < M) C[(size_t)row * ldc + col] = acc[i];
        }
    }
}

// ---------------------------------------------------------------------------
// Fused: partial[b] = sum over this block's tiles of |(Kitem*Vitem^T) - S|
// (M, N multiples of 16). Deterministic: partials reduced later in fixed order.
// ---------------------------------------------------------------------------
__global__ __launch_bounds__(256)
void ii_absdiff_wmma(const float* __restrict__ A, const float* __restrict__ B,
                     const float* __restrict__ S, float* __restrict__ partial,
                     int M, int N) {
    __shared__ float red[256];
    const int wave = threadIdx.x >> 5;
    const int lane = threadIdx.x & 31;
    const int tilesN = N >> 4;
    const int total = (M >> 4) * tilesN;
    int tile = blockIdx.x * 8 + wave;
    float lsum = 0.0f;
    if (tile < total) {
        int tm = tile / tilesN, tn = tile - tm * tilesN;
        int half = lane >> 4, l15 = lane & 15;
        const float* Ap = A + (size_t)(tm * 16 + l15) * 64 + half * 2;
        const float* Bp = B + (size_t)(tn * 16 + l15) * 64 + half * 2;
        __builtin_prefetch(S + (size_t)(tm * 16) * N + tn * 16, 0, 1);
        v8f acc = {};
#pragma unroll
        for (int k = 0; k < 64; k += 4) {
            v2f a; a.x = Ap[k]; a.y = Ap[k + 1];
            v2f b; b.x = Bp[k]; b.y = Bp[k + 1];
            acc = wmma_f32_16x16x4(a, b, acc);
        }
        int col = tn * 16 + l15;
#pragma unroll
        for (int i = 0; i < 8; ++i) {
            int row = tm * 16 + half * 8 + i;
            float d = acc[i] - S[(size_t)row * N + col];
            lsum += d < 0.0f ? -d : d;
        }
    }
    red[threadIdx.x] = lsum;
    __syncthreads();
    for (int s = 128; s > 0; s >>= 1) {
        if (threadIdx.x < s) red[threadIdx.x] += red[threadIdx.x + s];
        __syncthreads();
    }
    if (threadIdx.x == 0) partial[blockIdx.x] = red[0];
}

// ---------------------------------------------------------------------------
// Deterministic single-block sum: out = scale * sum(in[0..n))
// ---------------------------------------------------------------------------
__global__ __launch_bounds__(256)
void reduce_sum(const float* __restrict__ in, int n, float* __restrict__ out,
                float scale) {
    __shared__ float red[256];
    float a = 0.0f;
    for (int i = threadIdx.x; i < n; i += 256) a += in[i];
    red[threadIdx.x] = a;
    __syncthreads();
    for (int s = 128; s > 0; s >>= 1) {
        if (threadIdx.x < s) red[threadIdx.x] += red[threadIdx.x + s];
        __syncthreads();
    }
    if (threadIdx.x == 0) *out = red[0] * scale;
}

// ---------------------------------------------------------------------------
// contrast_loss: one block per row. Strided/affine view of the adjacency
// matrices (handles transposes, 1-(x>0)-eye, 1-x without materializing).
// ---------------------------------------------------------------------------
struct AdjDesc {
    const float* base;
    long rs;       // row stride (elements)
    long cs;       // col stride (elements)
    int  mode;     // 0: x   1: 1-(x>0)-eye(c+eyeoff==r)   2: 1-x
    int  eyeoff;
    int  C;        // number of columns
};

__device__ __forceinline__ float adj_val(const AdjDesc& d, int r, int c) {
    float x = d.base[(size_t)r * (size_t)d.rs + (size_t)c * (size_t)d.cs];
    if (d.mode == 1) {
        float v = 1.0f - (x > 0.0f ? 1.0f : 0.0f);
        if (c + d.eyeoff == r) v -= 1.0f;
        return v;
    }
    if (d.mode == 2) return 1.0f - x;
    return x;
}

__device__ __forceinline__ float u01_hash(unsigned a, unsigned b, unsigned c) {
    unsigned x = a * 0x9E3779B9u ^ b * 0x85EBCA6Bu ^ c * 0xC2B2AE35u;
    x ^= x >> 16; x *= 0x7FEB352Du;
    x ^= x >> 15; x *= 0x846CA68Bu;
    x ^= x >> 16;
    return (float)(x >> 8) * (1.0f / 16777216.0f);
}

#define CMAX 4096

__global__ __launch_bounds__(256)
void contrast_rows(AdjDesc P, AdjDesc Nd,
                   const float* __restrict__ PX, int pld,
                   const float* __restrict__ NX, int nld,
                   float* __restrict__ rowloss, unsigned seed) {
    __shared__ float s_vals[CMAX];
    __shared__ int   s_idx[CMAX];
    __shared__ float s_cdf[CMAX];
    __shared__ float s_redf[256];
    __shared__ int   s_redi[256];
    const int tid = threadIdx.x;
    const int r = blockIdx.x;

    // pcnt = sum(padj > 0)
    int cnt = 0;
    for (int c = tid; c < P.C; c += 256) cnt += (adj_val(P, r, c) > 0.0f) ? 1 : 0;
    s_redi[tid] = cnt; __syncthreads();
    for (int s = 128; s > 0; s >>= 1) { if (tid < s) s_redi[tid] += s_redi[tid + s]; __syncthreads(); }
    int pcnt = s_redi[0]; __syncthreads();

    // ncnt = sum(nadj > 0)
    cnt = 0;
    for (int c = tid; c < Nd.C; c += 256) cnt += (adj_val(Nd, r, c) > 0.0f) ? 1 : 0;
    s_redi[tid] = cnt; __syncthreads();
    for (int s = 128; s > 0; s >>= 1) { if (tid < s) s_redi[tid] += s_redi[tid + s]; __syncthreads(); }
    int ncnt = s_redi[0]; __syncthreads();

    int num = pcnt < ncnt ? pcnt : ncnt;
    if (num == 0) { if (tid == 0) rowloss[r] = 0.0f; return; }  // uniform branch
    bool branchA = (pcnt <= ncnt);   // num == pcnt -> sort padj, sample nadj

    // ---- ascending bitonic argsort of the "sorted" matrix row (pad +INF) ----
    const AdjDesc& Sd = branchA ? P : Nd;
    const int Cs = Sd.C;
    int C2 = 1; while (C2 < Cs) C2 <<= 1;
    for (int i = tid; i < C2; i += 256) {
        if (i < Cs) { s_vals[i] = adj_val(Sd, r, i); s_idx[i] = i; }
        else        { s_vals[i] = __builtin_inff(); s_idx[i] = Cs - 1; }
    }
    __syncthreads();
    for (int k = 2; k <= C2; k <<= 1) {
        for (int j = k >> 1; j > 0; j >>= 1) {
            for (int i = tid; i < C2; i += 256) {
                int ixj = i ^ j;
                if (ixj > i) {
                    bool up = ((i & k) == 0);
                    float v0 = s_vals[i], v1 = s_vals[ixj];
                    if (up ? (v0 > v1) : (v0 < v1)) {
                        s_vals[i] = v1; s_vals[ixj] = v0;
                        int t = s_idx[i]; s_idx[i] = s_idx[ixj]; s_idx[ixj] = t;
                    }
                }
            }
            __syncthreads();
        }
    }

    // ---- inclusive scan of relu(weights) for the sampled matrix ----
    const AdjDesc& Wd = branchA ? Nd : P;
    const int Cw = Wd.C;
    for (int i = tid; i < Cw; i += 256) {
        float w = adj_val(Wd, r, i);
        s_cdf[i] = w > 0.0f ? w : 0.0f;
    }
    __syncthreads();
    for (int off = 1; off < Cw; off <<= 1) {
        float t[16];           // CMAX/256 == 16
        int n = 0;
        for (int i = tid; i < Cw; i += 256, ++n) t[n] = (i >= off) ? s_cdf[i - off] : 0.0f;
        __syncthreads();
        n = 0;
        for (int i = tid; i < Cw; i += 256, ++n) s_cdf[i] += t[n];
        __syncthreads();
    }
    float total = s_cdf[Cw - 1];

    // ---- hinge terms over the first `num` draws ----
    float acc = 0.0f;
    for (int j = tid; j < num; j += 256) {
        float u = u01_hash(seed, (unsigned)r, (unsigned)j) * total;
        int lo = 0, hi = Cw;                       // searchsorted(side='right')
        while (lo < hi) { int mid = (lo + hi) >> 1; if (s_cdf[mid] <= u) lo = mid + 1; else hi = mid; }
        int sidx = lo < Cw ? lo : Cw - 1;
        int tidx = s_idx[Cs - num + j];            // top-num of ascending sort
        float nxv, pxv;
        if (branchA) { nxv = NX[(size_t)r * nld + sidx]; pxv = PX[(size_t)r * pld + tidx]; }
        else         { nxv = NX[(size_t)r * nld + tidx]; pxv = PX[(size_t)r * pld + sidx]; }
        float term = nxv - pxv + 1.0f;
        acc += term > 0.0f ? term : 0.0f;
    }
    s_redf[tid] = acc; __syncthreads();
    for (int s = 128; s > 0; s >>= 1) { if (tid < s) s_redf[tid] += s_redf[tid + s]; __syncthreads(); }
    if (tid == 0) rowloss[r] = s_redf[0] / (float)num;
}

// ---------------------------------------------------------------------------
extern "C" void kernel_launch(void* const* d_in, const int* in_sizes, int n_in,
                              void* d_out, int out_size, void* d_ws, size_t ws_size,
                              hipStream_t stream) {
    const float* KUU   = (const float*)d_in[0];   // [2051,2051]
    const float* VUU   = (const float*)d_in[1];   // [2048,2048]
    const float* Vuser = (const float*)d_in[2];   // [2048,64]
    const float* Vitem = (const float*)d_in[3];   // [4096,64]
    const float* Kuser = (const float*)d_in[4];   // [2051,64]
    const float* Kitem = (const float*)d_in[5];   // [4096,64]
    // d_in[6] = Vid (unused by the reference)
    const float* pref  = (const float*)d_in[7];   // [4096,5]
    const float* Vpref = (const float*)d_in[8];   // [4096,2048]
    const float* strct = (const float*)d_in[9];   // [4096,4096]
    float* out = (float*)d_out;                   // 5 scalars

    const int Nv = 2048, Nk = 2051, NvI = 4096, NkI = 4096;

    float* ws = (float*)d_ws;
    size_t o = 0;
    float* Vu_adj  = ws + o; o += (size_t)Nv * Nv;     // 2048x2048
    float* Ku_adj  = ws + o; o += (size_t)Nv * Nk;     // 2048x2051
    float* Vi_adj  = ws + o; o += (size_t)Nv * NvI;    // 2048x4096
    float* Ki_adj  = ws + o; o += (size_t)16 * NkI;    // 16x4096 (5 valid rows)
    float* rowloss = ws + o; o += 2048;
    float* iipart  = ws + o;                           // ii partials

    auto tiles = [](int M, int N) { return (size_t)((M + 15) / 16) * ((N + 15) / 16); };

    // ---- WMMA GEMMs ----
    gemm_nt_wmma<<<(unsigned)((tiles(Nv, Nv)  + 7) / 8), 256, 0, stream>>>(Vuser, Vuser, Vu_adj, Nv, Nv, Nv);
    gemm_nt_wmma<<<(unsigned)((tiles(Nv, Nk)  + 7) / 8), 256, 0, stream>>>(Kuser, Kuser, Ku_adj, Nv, Nk, Nk);
    gemm_nt_wmma<<<(unsigned)((tiles(Nv, NvI) + 7) / 8), 256, 0, stream>>>(Vuser, Vitem, Vi_adj, Nv, NvI, NvI);
    gemm_nt_wmma<<<(unsigned)((tiles(5,  NkI) + 7) / 8), 256, 0, stream>>>(Kuser, Kitem, Ki_adj, 5, NkI, NkI);

    // ---- ii = mean |Kitem*Vitem^T - structure| (fused WMMA + reduction) ----
    int iiBlocks = (int)((tiles(NkI, NvI) + 7) / 8);
    ii_absdiff_wmma<<<iiBlocks, 256, 0, stream>>>(Kitem, Vitem, strct, iipart, NkI, NvI);
    reduce_sum<<<1, 256, 0, stream>>>(iipart, iiBlocks, out + 4, 1.0f / ((float)NkI * (float)NvI));

    // ---- Vu: padj=VUU, nadj=1-(VUU>0)-eye, px=nx=Vu_adj ----
    AdjDesc P0{VUU, Nv, 1, 0, 0, Nv};
    AdjDesc N0{VUU, Nv, 1, 1, 0, Nv};
    contrast_rows<<<Nv, 256, 0, stream>>>(P0, N0, Vu_adj, Nv, Vu_adj, Nv, rowloss, 0xC0A10001u);
    reduce_sum<<<1, 256, 0, stream>>>(rowloss, Nv, out + 0, 1.0f);

    // ---- Ku: padj=KUU[:2048,:2048], nadj=(1-(KUU>0)-eye)[:2048,-3:] ----
    AdjDesc P1{KUU, Nk, 1, 0, 0, Nv};
    AdjDesc N1{KUU + (Nk - 3), Nk, 1, 1, Nk - 3, 3};
    contrast_rows<<<Nv, 256, 0, stream>>>(P1, N1, Ku_adj, Nk, Ku_adj + (Nk - 3), Nk, rowloss, 0xC0A10002u);
    reduce_sum<<<1, 256, 0, stream>>>(rowloss, Nv, out + 1, 1.0f);

    // ---- Vi: padj=Vpreference.T, nadj=1-Vpreference.T, px=nx=Vi_adj ----
    AdjDesc P2{Vpref, 1, Nv, 0, 0, NvI};
    AdjDesc N2{Vpref, 1, Nv, 2, 0, NvI};
    contrast_rows<<<Nv, 256, 0, stream>>>(P2, N2, Vi_adj, NvI, Vi_adj, NvI, rowloss, 0xC0A10003u);
    reduce_sum<<<1, 256, 0, stream>>>(rowloss, Nv, out + 2, 1.0f);

    // ---- Ki: padj=preference.T [5,4096], nadj=1-preference.T, px=nx=Ki_adj ----
    AdjDesc P3{pref, 1, 5, 0, 0, NkI};
    AdjDesc N3{pref, 1, 5, 2, 0, NkI};
    contrast_rows<<<5, 256, 0, stream>>>(P3, N3, Ki_adj, NkI, Ki_adj, NkI, rowloss, 0xC0A10004u);
    reduce_sum<<<1, 256, 0, stream>>>(rowloss, 5, out + 3, 1.0f);
}